// OptNetLatent_13924283974424
// MI455X (gfx1250) — compile-verified
//
#include <hip/hip_runtime.h>

typedef float v2f __attribute__((ext_vector_type(2)));
typedef float v8f __attribute__((ext_vector_type(8)));
typedef unsigned uv4 __attribute__((ext_vector_type(4)));
typedef int iv4 __attribute__((ext_vector_type(4)));
typedef int iv8 __attribute__((ext_vector_type(8)));

#define NX 729
#define NL 128
#define NM 256
#define K2 128          // NM/2 packed k-pair rows
#define PS 144          // padded stride of Gp rows, in float2 units (PS*8 % 256 == 128)
#define SM 132          // padded stride for M rows (floats)
#define N_ITERS 20
#define QPEN 0.1f
#define SIG 0.1f
#define BIGF 1e10f

// LDS layout (floats)
#define GPF (2*K2*PS)            // 36864
#define OFF_ML   GPF
#define OFF_VEC  (OFF_ML + NL*SM)
#define LDS_FLOATS (GPF + NL*SM + 10*NM + 4*NL + 732)

#if __has_builtin(__builtin_amdgcn_tensor_load_to_lds) && __has_builtin(__builtin_amdgcn_s_wait_tensorcnt)
#define USE_TDM 1
#else
#define USE_TDM 0
#endif

__global__ __launch_bounds__(256)
void optnet_qp(const float* __restrict__ puzzles,
               const float* __restrict__ Wi,
               const float* __restrict__ bi,
               const float* __restrict__ Gg,
               const float* __restrict__ zp,
               const float* __restrict__ sp,
               const float* __restrict__ Wo,
               const float* __restrict__ bo,
               float* __restrict__ out)
{
  extern __shared__ float smem[];
  v2f*   Gp   = (v2f*)smem;        // [K2][PS] packed pairs {G[2q][j], G[2q+1][j]}
  float* Ml   = smem + OFF_ML;     // [NL][SM]
  float* hv   = smem + OFF_VEC;    // [NM]
  float* pv   = hv + NM;           // [NL]
  float* zv   = pv + NL;           // [NL]
  float* sv   = zv + NL;           // [NM]
  float* lamv = sv + NM;           // [NM]
  float* rpv  = lamv + NM;         // [NM]
  float* rcv  = rpv + NM;          // [NM]
  float* Dv   = rcv + NM;          // [NM]
  float* gvv  = Dv + NM;           // [NM]
  float* rdv  = gvv + NM;          // [NL]
  float* rhs  = rdv + NL;          // [NL]
  float* dsv  = rhs + NL;          // [NM]
  float* dlv  = dsv + NM;          // [NM]
  float* red  = dlv + NM;          // [NM]
  float* xl   = red + NM;          // [732] puzzle row staging

  const int tid = threadIdx.x;
  const int b   = blockIdx.x;
  const float* xr = puzzles + (size_t)b * NX;

#if USE_TDM
  // Wave 0 kicks off a TDM 1-D tile load of this batch's puzzle row into LDS,
  // overlapped with the cooperative G staging below.
  if (tid < 32) {
    unsigned ldsoff = (unsigned)((const char*)xl - (const char*)smem);
    unsigned long long ga = (unsigned long long)(uintptr_t)xr;
    uv4 g0;
    g0.x = 1u;                                            // count=1, no gather
    g0.y = ldsoff;                                        // lds_addr (bytes)
    g0.z = (unsigned)ga;                                  // global_addr lo
    g0.w = (unsigned)((ga >> 32) & 0x01FFFFFFull) | (2u << 30); // addr hi | type=2
    iv8 g1;
    g1[0] = (int)(2u << 16);                              // wg_mask=0, data_size=4B
    g1[1] = (int)((NX & 0xFFFFu) << 16);                  // tensor_dim0 lo16
    g1[2] = (int)((NX >> 16) | (1u << 16));               // tensor_dim0 hi | tensor_dim1=1
    g1[3] = (int)(((unsigned)NX) << 16);                  // tile_dim0 = 729
    g1[4] = (int)1;                                       // tile_dim1=1, tile_dim2=0
    g1[5] = (int)NX;                                      // tensor_dim0_stride lo32
    g1[6] = 0;
    g1[7] = 0;
    iv4 g2 = {0,0,0,0}, g3 = {0,0,0,0};
#if defined(__clang_major__) && (__clang_major__ >= 23)
    iv8 g4 = {0,0,0,0,0,0,0,0};
    __builtin_amdgcn_tensor_load_to_lds(g0, g1, g2, g3, g4, 0);
#else
    __builtin_amdgcn_tensor_load_to_lds(g0, g1, g2, g3, 0);
#endif
  }
#else
  for (int k = tid; k < NX; k += 256) xl[k] = xr[k];
#endif

  // Stage G into LDS, k-pair interleaved: Gp[q][j] = {G[2q][j], G[2q+1][j]}
  for (int idx = tid; idx < K2*NL; idx += 256) {
    int q = idx >> 7, j = idx & 127;
    v2f g2v;
    g2v.x = Gg[(2*q)   * NL + j];
    g2v.y = Gg[(2*q+1) * NL + j];
    Gp[(size_t)q*PS + j] = g2v;
  }
  __syncthreads();

  // h = G z_p + s_p ; init s, lam  (thread t handles rows 2t, 2t+1)
  if (tid < K2) {
    const v2f* grow = Gp + (size_t)tid*PS;
    float a0 = sp[2*tid], a1 = sp[2*tid+1];
    for (int l = 0; l < NL; ++l) { v2f g = grow[l]; a0 += g.x*zp[l]; a1 += g.y*zp[l]; }
    hv[2*tid] = a0; hv[2*tid+1] = a1;
    sv[2*tid] = 1.0f; sv[2*tid+1] = 1.0f;
    lamv[2*tid] = 1.0f; lamv[2*tid+1] = 1.0f;
  }
#if USE_TDM
  if (tid < 32) __builtin_amdgcn_s_wait_tensorcnt(0);
#endif
  __syncthreads();  // Gp, xl, h ready

  if (tid < NL) { // p = x Wi^T + bi ; z = 0
    const float* wr = Wi + (size_t)tid * NX;
    float acc = bi[tid];
    for (int k = 0; k < NX; ++k) acc += xl[k] * wr[k];
    pv[tid] = acc;
    zv[tid] = 0.0f;
  }
  __syncthreads();

  const int lane  = tid & 31;
  const int wave  = tid >> 5;
  const int mrow  = lane & 15;
  const int g     = lane >> 4;
  const int rbase = wave << 4;

  for (int it = 0; it < N_ITERS; ++it) {
    // rp[m] = z.G^T + s - h  (paired rows per thread)
    if (tid < K2) {
      const v2f* grow = Gp + (size_t)tid*PS;
      float a0 = 0.0f, a1 = 0.0f;
      for (int i = 0; i < NL; ++i) { v2f gg = grow[i]; a0 += zv[i]*gg.x; a1 += zv[i]*gg.y; }
      int m0 = 2*tid, m1 = m0+1;
      rpv[m0] = sv[m0] - hv[m0] + a0;
      rpv[m1] = sv[m1] - hv[m1] + a1;
    }
    // rd[i] = 0.1 z + p + lam.G  (column read: one b64 covers lam pair)
    if (tid < NL) {
      float acc = QPEN * zv[tid] + pv[tid];
      for (int q = 0; q < K2; ++q) {
        v2f gg = Gp[(size_t)q*PS + tid];
        acc += lamv[2*q]*gg.x + lamv[2*q+1]*gg.y;
      }
      rdv[tid] = acc;
    }
    // mu = mean(lam*s)
    red[tid] = lamv[tid] * sv[tid];
    __syncthreads();
    for (int off = 128; off > 0; off >>= 1) {
      if (tid < off) red[tid] += red[tid + off];
      __syncthreads();
    }
    float mu = red[0] * (1.0f / NM);
    {
      float l = lamv[tid], s0 = sv[tid];
      float rc = l * s0 - SIG * mu;
      rcv[tid] = rc;
      Dv[tid]  = l / s0;
      gvv[tid] = (-rc + l * rpv[tid]) / s0;
    }
    __syncthreads();
    if (tid < NL) {
      float acc = rdv[tid];
      for (int q = 0; q < K2; ++q) {
        v2f gg = Gp[(size_t)q*PS + tid];
        acc += gvv[2*q]*gg.x + gvv[2*q+1]*gg.y;
      }
      rhs[tid] = -acc;
    }

    // ---- M = 0.1 I + G^T diag(D) G via V_WMMA_F32_16X16X4_F32 ----
    // Wave w owns rows [16w,16w+16); packed layout: one ds_load_b64 per fragment.
    v8f acc[8];
    const v8f z8 = {0,0,0,0,0,0,0,0};
#pragma unroll
    for (int t = 0; t < 8; ++t) acc[t] = z8;

    for (int q2 = 0; q2 < K2; q2 += 2) {
      int q = q2 + g;                        // lanes 0-15: K=2q2,2q2+1; 16-31: +2
      v2f dd = *(const v2f*)&Dv[2*q];
      v2f ga = Gp[(size_t)q*PS + rbase + mrow];
      v2f A  = ga * dd;                      // v_pk_mul_f32
#pragma unroll
      for (int t = 0; t < 8; ++t) {
        v2f Bf = Gp[(size_t)q*PS + t*16 + mrow];
        acc[t] = __builtin_amdgcn_wmma_f32_16x16x4_f32(
            false, A, false, Bf, (short)0, acc[t], false, false);
      }
    }
#pragma unroll
    for (int t = 0; t < 8; ++t) {
#pragma unroll
      for (int r = 0; r < 8; ++r) {
        int row = rbase + r + 8*g;           // C/D layout: VGPR r -> M=r (+8 hi lanes)
        int col = t*16 + mrow;
        float v = acc[t][r];
        if (row == col) v += QPEN;
        Ml[row*SM + col] = v;
      }
    }
    __syncthreads();

    // ---- Cholesky factorization (lower triangle, in place) ----
    for (int j = 0; j < NL; ++j) {
      if (tid == 0) {
        float l = sqrtf(Ml[j*SM + j]);
        Ml[j*SM + j] = l;
        red[0] = 1.0f / l;
      }
      __syncthreads();
      float inv = red[0];
      int i = j + 1 + tid;
      if (i < NL) Ml[i*SM + j] *= inv;
      __syncthreads();
      if (i < NL) {
        float lij = Ml[i*SM + j];
        for (int k = j + 1; k <= i; ++k)
          Ml[i*SM + k] -= lij * Ml[k*SM + j];
      }
      __syncthreads();
    }

    // Forward solve L y = rhs
    for (int j = 0; j < NL; ++j) {
      if (tid == 0) rhs[j] /= Ml[j*SM + j];
      __syncthreads();
      float yj = rhs[j];
      int i = j + 1 + tid;
      if (i < NL) rhs[i] -= Ml[i*SM + j] * yj;
      __syncthreads();
    }
    // Backward solve L^T dz = y (rhs becomes dz)
    for (int j = NL - 1; j >= 0; --j) {
      if (tid == 0) rhs[j] /= Ml[j*SM + j];
      __syncthreads();
      float zj = rhs[j];
      if (tid < j) rhs[tid] -= Ml[j*SM + tid] * zj;   // L^T[i][j] = L[j][i]
      __syncthreads();
    }

    // ds, dlam, fraction-to-boundary (paired rows per thread)
    if (tid < K2) {
      const v2f* grow = Gp + (size_t)tid*PS;
      float a0 = 0.0f, a1 = 0.0f;
      for (int i = 0; i < NL; ++i) { v2f gg = grow[i]; a0 += rhs[i]*gg.x; a1 += rhs[i]*gg.y; }
      int m0 = 2*tid, m1 = m0+1;
      float ds0 = -rpv[m0] - a0, ds1 = -rpv[m1] - a1;
      dsv[m0] = ds0; dsv[m1] = ds1;
      float dl0 = (-rcv[m0] - lamv[m0]*ds0) / sv[m0];
      float dl1 = (-rcv[m1] - lamv[m1]*ds1) / sv[m1];
      dlv[m0] = dl0; dlv[m1] = dl1;
      float c = BIGF;
      c = fminf(c, (ds0 < 0.0f) ? (-sv[m0]/ds0)   : BIGF);
      c = fminf(c, (ds1 < 0.0f) ? (-sv[m1]/ds1)   : BIGF);
      c = fminf(c, (dl0 < 0.0f) ? (-lamv[m0]/dl0) : BIGF);
      c = fminf(c, (dl1 < 0.0f) ? (-lamv[m1]/dl1) : BIGF);
      red[tid] = c;
    }
    __syncthreads();
    for (int off = 64; off > 0; off >>= 1) {
      if (tid < off) red[tid] = fminf(red[tid], red[tid + off]);
      __syncthreads();
    }
    float alpha = fminf(1.0f, 0.99f * red[0]);
    sv[tid]   += alpha * dsv[tid];
    lamv[tid] += alpha * dlv[tid];
    if (tid < NL) zv[tid] += alpha * rhs[tid];
    __syncthreads();
  }

  // out = z* Wo^T + bo
  for (int j = tid; j < NX; j += 256) {
    const float* wr = Wo + (size_t)j * NL;
    float acc = bo[j];
    for (int l = 0; l < NL; ++l) acc += zv[l] * wr[l];
    out[(size_t)b * NX + j] = acc;
  }
}

extern "C" void kernel_launch(void* const* d_in, const int* in_sizes, int n_in,
                              void* d_out, int out_size, void* d_ws, size_t ws_size,
                              hipStream_t stream) {
  const float* puzzles = (const float*)d_in[0];
  const float* Wi = (const float*)d_in[1];
  const float* bi = (const float*)d_in[2];
  const float* Gg = (const float*)d_in[3];
  const float* zp = (const float*)d_in[4];
  const float* sp = (const float*)d_in[5];
  const float* Wo = (const float*)d_in[6];
  const float* bo = (const float*)d_in[7];
  float* out = (float*)d_out;
  (void)d_ws; (void)ws_size; (void)n_in; (void)out_size;

  int B = in_sizes[0] / NX;   // 1024
  size_t lds_bytes = (size_t)LDS_FLOATS * sizeof(float); // ~230 KB, < 320 KB/WGP
  hipLaunchKernelGGL(optnet_qp, dim3(B), dim3(256), lds_bytes, stream,
                     puzzles, Wi, bi, Gg, zp, sp, Wo, bo, out);
}